// SpectralConv3d_UCFT_51376398795177
// MI455X (gfx1250) — compile-verified
//
#include <hip/hip_runtime.h>
#include <math.h>

// ---------------------------------------------------------------------------
// SpectralConv3d via truncated DCT, all contractions on V_WMMA_F32_16X16X4_F32
// B=8, CI=CO=32, S=64^3, modes=16^3.
//
// Roofline: ~5.7 GFLOP fp32 vs 268 MB in + 268 MB out mandatory HBM traffic
// -> memory bound (~23 us @ 23.3 TB/s). Intermediates (~92 MB) are kept
// L2-resident (192 MB L2); the streaming input/output use non-temporal
// cache hints (th:NT) so they don't evict the intermediates.
//
// Workspace layout (floats):
//   ct  [16*64]                      : truncated DCT matrix C[:16,:]
//   T1  [8*32*64*64*16] = 16,777,216 : after Z-DCT   (reused as T6)
//   T2  [8*32*64*16*16] =  4,194,304 : after Y-DCT   (reused as T5)
//   T3  [8*32*16*16*16] =  1,048,576 : after X-DCT
//   T4  [8*32*16*16*16] =  1,048,576 : after channel mix
// ---------------------------------------------------------------------------

typedef __attribute__((ext_vector_type(2))) float v2f;
typedef __attribute__((ext_vector_type(8))) float v8f;

static __device__ __forceinline__ v8f wmma4(v2f a, v2f b, v8f c) {
  // D(16x16,f32) = A(16x4,f32) * B(4x16,f32) + C
  return __builtin_amdgcn_wmma_f32_16x16x4_f32(
      /*neg_a=*/false, a, /*neg_b=*/false, b,
      /*c_mod=*/(short)0, c, /*reuse_a=*/false, /*reuse_b=*/false);
}

// Truncated DCT-II matrix: ct[k][n] = s * cos(pi*(n+0.5)*k/64), k<16, n<64
__global__ void k_init_ct(float* __restrict__ ct) {
  int t = blockIdx.x * blockDim.x + threadIdx.x;
  if (t < 16 * 64) {
    int k = t >> 6, n = t & 63;
    double v = sqrt(2.0 / 64.0) * cos(M_PI * (n + 0.5) * (double)k / 64.0);
    if (k == 0) v *= (1.0 / sqrt(2.0));
    ct[t] = (float)v;
  }
}

// Stage A: T1[b,c,x,y,kz] = sum_z X[b,c,x,y,z] * Ct[kz,z]
// One wave = one (b,c,x,ytile) 16x16 output tile, K=64 in 16 steps.
// X is a one-shot 268 MB stream -> non-temporal loads (don't pollute L2).
__global__ __launch_bounds__(256) void k_dct_z(const float* __restrict__ X,
                                               const float* __restrict__ ct,
                                               float* __restrict__ T1) {
  const unsigned w    = (blockIdx.x * blockDim.x + threadIdx.x) >> 5;
  const unsigned lane = threadIdx.x & 31;
  const unsigned ln = lane & 15, hi = lane >> 4, kb = hi * 2;

  const unsigned ytile = w & 3;
  const unsigned x     = (w >> 2) & 63;
  const unsigned bc    = w >> 8;          // b*32+c
  const unsigned ybase = ytile * 16;

  const float* src = X + ((size_t)bc * 64 + x) * 4096;   // (y=64, z=64)
  v8f acc = {};
#pragma unroll 4
  for (int kg = 0; kg < 64; kg += 4) {
    v2f a = __builtin_nontemporal_load(
        (const v2f*)(src + (ybase + ln) * 64 + kg + kb));     // A[y][z]
    v2f b = *(const v2f*)(ct + ln * 64 + kg + kb);            // B[z][kz]=Ct[kz][z]
    acc = wmma4(a, b, acc);
  }
  float* dst = T1 + ((size_t)bc * 64 + x) * 1024;            // (y=64, kz=16)
#pragma unroll
  for (int r = 0; r < 8; ++r)
    dst[(ybase + r + hi * 8) * 16 + ln] = acc[r];
}

// Stage B: T2[b,c,x,ky,kz] = sum_y Ct[ky,y] * T1[b,c,x,y,kz]
__global__ __launch_bounds__(256) void k_dct_y(const float* __restrict__ T1,
                                               const float* __restrict__ ct,
                                               float* __restrict__ T2) {
  const unsigned w    = (blockIdx.x * blockDim.x + threadIdx.x) >> 5;
  const unsigned lane = threadIdx.x & 31;
  const unsigned ln = lane & 15, hi = lane >> 4, kb = hi * 2;

  const float* src = T1 + (size_t)w * 1024;                  // (y=64, kz=16)
  v8f acc = {};
#pragma unroll 4
  for (int kg = 0; kg < 64; kg += 4) {
    v2f a = *(const v2f*)(ct + ln * 64 + kg + kb);           // A[ky][y]=Ct
    v2f b;
    b.x = src[(kg + kb) * 16 + ln];                          // B[y][kz]
    b.y = src[(kg + kb + 1) * 16 + ln];
    acc = wmma4(a, b, acc);
  }
  float* dst = T2 + (size_t)w * 256;                         // (ky=16, kz=16)
#pragma unroll
  for (int r = 0; r < 8; ++r)
    dst[(r + hi * 8) * 16 + ln] = acc[r];
}

// Stage C: T3[b,c,kx,n] = sum_x Ct[kx,x] * T2[b,c,x,n],  n=(ky,kz) 0..255
__global__ __launch_bounds__(256) void k_dct_x(const float* __restrict__ T2,
                                               const float* __restrict__ ct,
                                               float* __restrict__ T3) {
  const unsigned w    = (blockIdx.x * blockDim.x + threadIdx.x) >> 5;
  const unsigned lane = threadIdx.x & 31;
  const unsigned ln = lane & 15, hi = lane >> 4, kb = hi * 2;

  const unsigned nt = w & 15;
  const unsigned bc = w >> 4;
  const float* src = T2 + (size_t)bc * 16384;                // (x=64, n=256)
  v8f acc = {};
#pragma unroll 4
  for (int kg = 0; kg < 64; kg += 4) {
    v2f a = *(const v2f*)(ct + ln * 64 + kg + kb);
    v2f b;
    b.x = src[(kg + kb) * 256 + nt * 16 + ln];
    b.y = src[(kg + kb + 1) * 256 + nt * 16 + ln];
    acc = wmma4(a, b, acc);
  }
  float* dst = T3 + (size_t)bc * 4096;                       // (kx=16, n=256)
#pragma unroll
  for (int r = 0; r < 8; ++r)
    dst[(r + hi * 8) * 256 + nt * 16 + ln] = acc[r];
}

// Stage D: T4[b,o,m] = sum_i T3[b,i,m] * W[i,o,m]   per mode m (4096 modes)
// M=8 valid rows (b), padded to 16; N=32 in two otiles; K=32 in 8 steps.
__global__ __launch_bounds__(256) void k_mix(const float* __restrict__ T3,
                                             const float* __restrict__ W,
                                             float* __restrict__ T4) {
  const unsigned w    = (blockIdx.x * blockDim.x + threadIdx.x) >> 5;
  const unsigned lane = threadIdx.x & 31;
  const unsigned ln = lane & 15, hi = lane >> 4, kb = hi * 2;

  const unsigned ot = w & 1;
  const unsigned m  = w >> 1;
  const unsigned arow = ln & 7;     // duplicate rows 8..15 -> safe loads
  v8f acc = {};
#pragma unroll
  for (int kg = 0; kg < 32; kg += 4) {
    v2f a, b;
    a.x = T3[(size_t)(arow * 32 + kg + kb) * 4096 + m];
    a.y = T3[(size_t)(arow * 32 + kg + kb + 1) * 4096 + m];
    b.x = W[(size_t)((kg + kb) * 32 + ot * 16 + ln) * 4096 + m];
    b.y = W[(size_t)((kg + kb + 1) * 32 + ot * 16 + ln) * 4096 + m];
    acc = wmma4(a, b, acc);
  }
  if (hi == 0) {                    // only rows 0..7 (b) are real
#pragma unroll
    for (int r = 0; r < 8; ++r)
      T4[(size_t)(r * 32 + ot * 16 + ln) * 4096 + m] = acc[r];
  }
}

// Stage E: T5[b,o,X,n] = sum_kx Ct[kx,X] * T4[b,o,kx,n]
__global__ __launch_bounds__(256) void k_idct_x(const float* __restrict__ T4,
                                                const float* __restrict__ ct,
                                                float* __restrict__ T5) {
  const unsigned w    = (blockIdx.x * blockDim.x + threadIdx.x) >> 5;
  const unsigned lane = threadIdx.x & 31;
  const unsigned ln = lane & 15, hi = lane >> 4, kb = hi * 2;

  const unsigned nt = w & 15;
  const unsigned mt = (w >> 4) & 3;
  const unsigned bo = w >> 6;
  const float* src = T4 + (size_t)bo * 4096;                 // (kx=16, n=256)
  v8f acc = {};
#pragma unroll
  for (int kg = 0; kg < 16; kg += 4) {
    v2f a, b;
    a.x = ct[(kg + kb) * 64 + mt * 16 + ln];                 // A[X][kx]=Ct^T
    a.y = ct[(kg + kb + 1) * 64 + mt * 16 + ln];
    b.x = src[(kg + kb) * 256 + nt * 16 + ln];
    b.y = src[(kg + kb + 1) * 256 + nt * 16 + ln];
    acc = wmma4(a, b, acc);
  }
  float* dst = T5 + (size_t)bo * 16384;                      // (X=64, n=256)
#pragma unroll
  for (int r = 0; r < 8; ++r)
    dst[(mt * 16 + r + hi * 8) * 256 + nt * 16 + ln] = acc[r];
}

// Stage F: T6[b,o,X,Y,kz] = sum_ky Ct[ky,Y] * T5[b,o,X,ky,kz]
__global__ __launch_bounds__(256) void k_idct_y(const float* __restrict__ T5,
                                                const float* __restrict__ ct,
                                                float* __restrict__ T6) {
  const unsigned w    = (blockIdx.x * blockDim.x + threadIdx.x) >> 5;
  const unsigned lane = threadIdx.x & 31;
  const unsigned ln = lane & 15, hi = lane >> 4, kb = hi * 2;

  const unsigned mt = w & 3;
  const unsigned x  = (w >> 2) & 63;
  const unsigned bo = w >> 8;
  const float* src = T5 + (size_t)bo * 16384 + x * 256;      // (ky=16, kz=16)
  v8f acc = {};
#pragma unroll
  for (int kg = 0; kg < 16; kg += 4) {
    v2f a, b;
    a.x = ct[(kg + kb) * 64 + mt * 16 + ln];                 // A[Y][ky]=Ct^T
    a.y = ct[(kg + kb + 1) * 64 + mt * 16 + ln];
    b.x = src[(kg + kb) * 16 + ln];
    b.y = src[(kg + kb + 1) * 16 + ln];
    acc = wmma4(a, b, acc);
  }
  float* dst = T6 + ((size_t)bo * 64 + x) * 1024;            // (Y=64, kz=16)
#pragma unroll
  for (int r = 0; r < 8; ++r)
    dst[(mt * 16 + r + hi * 8) * 16 + ln] = acc[r];
}

// Stage G: out[b,o,X,Y,Z] = sum_kz T6[b,o,X,Y,kz] * Ct[kz,Z]
// out is a one-shot 268 MB stream -> non-temporal stores (don't pollute L2).
__global__ __launch_bounds__(256) void k_idct_z(const float* __restrict__ T6,
                                                const float* __restrict__ ct,
                                                float* __restrict__ out) {
  const unsigned w    = (blockIdx.x * blockDim.x + threadIdx.x) >> 5;
  const unsigned lane = threadIdx.x & 31;
  const unsigned ln = lane & 15, hi = lane >> 4, kb = hi * 2;

  const unsigned nt = w & 3;
  const unsigned mt = (w >> 2) & 3;
  const unsigned x  = (w >> 4) & 63;
  const unsigned bo = w >> 10;
  const float* src = T6 + ((size_t)bo * 64 + x) * 1024;      // (Y=64, kz=16)
  v8f acc = {};
#pragma unroll
  for (int kg = 0; kg < 16; kg += 4) {
    v2f a = *(const v2f*)(src + (mt * 16 + ln) * 16 + kg + kb); // A[Y][kz]
    v2f b;
    b.x = ct[(kg + kb) * 64 + nt * 16 + ln];                 // B[kz][Z]=Ct
    b.y = ct[(kg + kb + 1) * 64 + nt * 16 + ln];
    acc = wmma4(a, b, acc);
  }
  float* dst = out + ((size_t)bo * 64 + x) * 4096;           // (Y=64, Z=64)
#pragma unroll
  for (int r = 0; r < 8; ++r)
    __builtin_nontemporal_store(
        acc[r], dst + (mt * 16 + r + hi * 8) * 64 + nt * 16 + ln);
}

extern "C" void kernel_launch(void* const* d_in, const int* in_sizes, int n_in,
                              void* d_out, int out_size, void* d_ws, size_t ws_size,
                              hipStream_t stream) {
  const float* x   = (const float*)d_in[0];   // (8,32,64,64,64)
  const float* wgt = (const float*)d_in[1];   // (32,32,16,16,16)
  float* out = (float*)d_out;                 // (8,32,64,64,64)

  float* ws = (float*)d_ws;
  float* ct = ws;                       // 1,024
  float* T1 = ws + 1024;                // 16,777,216
  float* T2 = T1 + 16777216;            //  4,194,304
  float* T3 = T2 + 4194304;             //  1,048,576
  float* T4 = T3 + 1048576;             //  1,048,576
  float* T5 = T2;                       // reuse (T2 dead after stage C)
  float* T6 = T1;                       // reuse (T1 dead after stage B)

  k_init_ct<<<4,     256, 0, stream>>>(ct);
  k_dct_z  <<<8192,  256, 0, stream>>>(x,  ct, T1);   // 65536 waves
  k_dct_y  <<<2048,  256, 0, stream>>>(T1, ct, T2);   // 16384 waves
  k_dct_x  <<<512,   256, 0, stream>>>(T2, ct, T3);   //  4096 waves
  k_mix    <<<1024,  256, 0, stream>>>(T3, wgt, T4);  //  8192 waves
  k_idct_x <<<2048,  256, 0, stream>>>(T4, ct, T5);   // 16384 waves
  k_idct_y <<<8192,  256, 0, stream>>>(T5, ct, T6);   // 65536 waves
  k_idct_z <<<32768, 256, 0, stream>>>(T6, ct, out);  // 262144 waves
}